// RouteChoiceBase_35605278884489
// MI455X (gfx1250) — compile-verified
//
#include <hip/hip_runtime.h>
#include <hip/hip_bf16.h>

// ---------------------------------------------------------------------------
// RouteChoice on MI455X (gfx1250, wave32).
//   - Structure-exploited solve: Z[d,:2000] = (I - Mc)^{-1} B_conn[d,:]^T,
//     Z[d,2000] = 1.  Jacobi/Neumann iteration X <- C + Mc*X, 256 steps.
//   - Mc (16 MB fp32) lives in workspace and stays L2-resident (192 MB L2).
//   - GEMM step uses V_WMMA_F32_16X16X4_F32 (full fp32, wave32 WMMA) with
//     4-way split-K per 16-row tile + LDS reduction (shortens the latency-
//     bound critical path of each of the 256 dependent Jacobi launches).
// ---------------------------------------------------------------------------

#if defined(__HIP_DEVICE_COMPILE__) && defined(__AMDGCN__)
#if !__has_builtin(__builtin_amdgcn_wmma_f32_16x16x4_f32)
#error "missing __builtin_amdgcn_wmma_f32_16x16x4_f32 on this toolchain"
#endif
#endif

typedef float v2f __attribute__((ext_vector_type(2)));
typedef float v8f __attribute__((ext_vector_type(8)));

#define NLINK   2000
#define N1      2001
#define NDEST   16
#define NNZ     8000
#define NTILES  125           // 2000 / 16
#define KSPLIT  4
#define KSPAN   (NLINK / KSPLIT)   // 500 K-values per wave
#define JACOBI_ITERS 256

// ---- mu[i] = exp(feat . scale_w); dummy row features = (1,1,1) -------------
__global__ void rc_mu_kernel(const float* __restrict__ OL,
                             const float* __restrict__ TT,
                             const float* __restrict__ sw,
                             float* __restrict__ mu) {
  int i = blockIdx.x * blockDim.x + threadIdx.x;
  if (i >= N1) return;
  float f0, f1, f2;
  if (i < NLINK) { f0 = OL[i]; f1 = TT[i]; f2 = 0.0f; }
  else           { f0 = 1.0f;  f1 = 1.0f;  f2 = 1.0f; }
  mu[i] = expf(f0 * sw[0] + f1 * sw[1] + f2 * sw[2]);
}

// ---- per-edge reward; COO scatter-add into ireward_smat and M --------------
__global__ void rc_edge_kernel(const float* __restrict__ bf,
                               const int* __restrict__ src,
                               const int* __restrict__ dst,
                               const float* __restrict__ bw,
                               const float* __restrict__ mu,
                               float* __restrict__ irs,   // [2000*2000] out
                               float* __restrict__ M) {   // [2000*2000] ws
  int e = blockIdx.x * blockDim.x + threadIdx.x;
  if (e >= NNZ) return;
  const float* f = bf + (size_t)e * 4;
  float ir = f[0] * bw[0] + f[1] * bw[1] + f[2] * bw[2] + f[3] * bw[3];
  int s = src[e], d = dst[e];
  size_t idx = (size_t)s * NLINK + d;
  atomicAdd(&irs[idx], ir);
  atomicAdd(&M[idx], expf(ir / mu[s]));
}

// ---- C[j][d] = B_conn[d][j]; X0 = C (first Jacobi iterate) -----------------
__global__ void rc_initC_kernel(const float* __restrict__ Bc,
                                float* __restrict__ C,
                                float* __restrict__ X0) {
  int idx = blockIdx.x * blockDim.x + threadIdx.x;
  if (idx >= NLINK * NDEST) return;
  int j = idx >> 4, d = idx & 15;
  float v = Bc[(size_t)d * NLINK + j];
  C[idx] = v;
  X0[idx] = v;
}

// ---- Jacobi step: Xout = C + M * Xin  (f32 WMMA 16x16x4, split-K x4) -------
// One block (4 waves) per 16-row tile of M; each wave covers 500 K-values
// (125 WMMAs), partial 16x16 f32 accumulators reduced through LDS.
// A frag (16x4 f32, 2 VGPR): lanes 0-15 -> K0/K1, lanes 16-31 -> K2/K3,
//   row = lane&15.  B frag (4x16): lanes 0-15 -> rows K0/K1, lanes 16-31 ->
//   rows K2/K3, col = lane&15.  C/D: vgpr j = row j (+8 for upper half-wave).
__global__ __launch_bounds__(128, 1)
void rc_jacobi_kernel(const float* __restrict__ M,
                      const float* __restrict__ C,
                      const float* __restrict__ Xin,
                      float* __restrict__ Xout) {
  __shared__ float red[KSPLIT][8][32];
  int tile = blockIdx.x;            // 0..124
  int kw   = threadIdx.x >> 5;      // 0..3 : K-split slice
  int lane = threadIdx.x & 31;
  int half = lane >> 4;             // 0 or 1
  int lcol = lane & 15;
  int row0 = tile * 16;
  int k0   = kw * KSPAN;

  v8f acc = {};   // partial sum; C added after reduction

  const float* arow = M + (size_t)(row0 + lcol) * NLINK + k0 + 2 * half;
  const float* bcol = Xin + (size_t)(k0 + 2 * half) * NDEST + lcol;

#pragma unroll 5
  for (int k = 0; k < KSPAN; k += 4) {
    v2f a = *(const v2f*)(arow + k);                 // M[row][k0+k+2h .. +1]
    v2f b;
    b[0] = bcol[(size_t)k * NDEST];                  // X[k0+k+2h  ][lcol]
    b[1] = bcol[(size_t)k * NDEST + NDEST];          // X[k0+k+2h+1][lcol]
    acc = __builtin_amdgcn_wmma_f32_16x16x4_f32(
        /*neg_a=*/false, a, /*neg_b=*/false, b,
        /*c_mod=*/(short)0, acc, /*reuse_a=*/false, /*reuse_b=*/false);
  }

#pragma unroll
  for (int j = 0; j < 8; ++j) red[kw][j][lane] = acc[j];
  __syncthreads();

  if (kw == 0) {
#pragma unroll
    for (int j = 0; j < 8; ++j) {
      float s = red[0][j][lane] + red[1][j][lane]
              + red[2][j][lane] + red[3][j][lane];
      size_t o = (size_t)(row0 + j + 8 * half) * NDEST + lcol;
      Xout[o] = s + C[o];
    }
  }
}

// ---- Z[d][j] = X[j][d] (j<2000) else 1 ; V = mu * log(Z) -------------------
__global__ void rc_final_kernel(const float* __restrict__ X,
                                const float* __restrict__ mu,
                                float* __restrict__ V,
                                float* __restrict__ Z) {
  int idx = blockIdx.x * blockDim.x + threadIdx.x;
  if (idx >= NDEST * N1) return;
  int d = idx / N1, j = idx - d * N1;
  float z = (j < NLINK) ? X[(size_t)j * NDEST + d] : 1.0f;
  Z[idx] = z;
  V[idx] = mu[j] * logf(z);
}

// ---------------------------------------------------------------------------
extern "C" void kernel_launch(void* const* d_in, const int* in_sizes, int n_in,
                              void* d_out, int out_size, void* d_ws, size_t ws_size,
                              hipStream_t stream) {
  const float* beta_feat = (const float*)d_in[0];   // [8000,4]
  const float* OL_deg    = (const float*)d_in[1];   // [2000]
  const float* TT_mean   = (const float*)d_in[2];   // [2000]
  const float* B_conn    = (const float*)d_in[3];   // [16,2000]
  const float* beta_w    = (const float*)d_in[4];   // [1,4]
  const float* scale_w   = (const float*)d_in[5];   // [1,3]
  const int*   src       = (const int*)d_in[6];     // [8000]
  const int*   dst       = (const int*)d_in[7];     // [8000]

  float* out = (float*)d_out;
  float* irs = out;                                  // [2000*2000]
  float* mu  = out + (size_t)NLINK * NLINK;          // [2001]
  float* V   = mu + N1;                              // [16*2001]
  float* Z   = V + (size_t)NDEST * N1;               // [16*2001]

  float* ws = (float*)d_ws;
  float* M  = ws;                                    // [2000*2000]  16 MB
  float* C  = M + (size_t)NLINK * NLINK;             // [2000*16]
  float* X0 = C + (size_t)NLINK * NDEST;             // [2000*16]
  float* X1 = X0 + (size_t)NLINK * NDEST;            // [2000*16]

  (void)hipMemsetAsync(irs, 0, (size_t)NLINK * NLINK * sizeof(float), stream);
  (void)hipMemsetAsync(M,   0, (size_t)NLINK * NLINK * sizeof(float), stream);

  rc_mu_kernel<<<(N1 + 255) / 256, 256, 0, stream>>>(OL_deg, TT_mean, scale_w, mu);
  rc_edge_kernel<<<(NNZ + 255) / 256, 256, 0, stream>>>(beta_feat, src, dst,
                                                        beta_w, mu, irs, M);
  rc_initC_kernel<<<(NLINK * NDEST + 255) / 256, 256, 0, stream>>>(B_conn, C, X0);

  float* Xin = X0;
  float* Xout = X1;
  for (int it = 0; it < JACOBI_ITERS; ++it) {
    rc_jacobi_kernel<<<NTILES, 128, 0, stream>>>(M, C, Xin, Xout);
    float* t = Xin; Xin = Xout; Xout = t;
  }

  rc_final_kernel<<<(NDEST * N1 + 255) / 256, 256, 0, stream>>>(Xin, mu, V, Z);
}